// RegimeFeatureExtractor_81106162418314
// MI455X (gfx1250) — compile-verified
//
#include <hip/hip_runtime.h>
#include <hip/hip_bf16.h>
#include <math.h>
#include <stdint.h>

// ---------------- configuration ----------------
#define TPB     256          // threads per block (8 wave32 waves)
#define TILE    2048         // timesteps per block
#define CPT     8            // timesteps per thread (TILE / TPB)
#define HALO    21           // price lookback needed per output
#define LDPAD   32           // halo slots (chunk-aligned) staged before the tile
#define NLOAD   (TILE + LDPAD)     // 2080 floats of prices staged in LDS
#define NCH     (NLOAD / 4)        // 520 b128 chunks for the input DMA
#define T_LEN   8192
#define NF      6
#define EPSF    1e-8f
#define NW      (CPT + HALO)       // 29 local window slots per thread

__device__ __forceinline__ float std_from(float s, float q, float cnt) {
    // unbiased rolling std from sum / sum-of-squares, matching the reference
    float var = (q - s * s / cnt) / fmaxf(cnt - 1.0f, 1.0f);
    float sd  = sqrtf(fmaxf(var, 0.0f));
    return (cnt >= 2.0f) ? sd : 1e-6f;
}

__global__ __launch_bounds__(TPB)
void RegimeFeatureExtractor_81106162418314_kernel(const float* __restrict__ prices,
                                                  float* __restrict__ out)
{
    __shared__ __align__(16) float s_in[NLOAD];        // staged prices
    __shared__ __align__(16) float s_out[TILE * NF];   // staged features (48 KB)

    const int tid   = threadIdx.x;
    const int row   = blockIdx.x;
    const int tile0 = blockIdx.y * TILE;
    const long long row_base = (long long)row * T_LEN;
    const int gstart = tile0 - LDPAD;                  // global t of s_in[0]

    // ---- Phase A: async DMA prices[row, gstart .. gstart+NLOAD) into LDS ----
    {
        const float* rowp = prices + row_base;
        const unsigned lds_base = (unsigned)(uintptr_t)&s_in[0];
        for (int i = tid; i < NCH; i += TPB) {
            int g  = gstart + i * 4;                   // chunk-aligned (16B)
            int gc = (g < 0) ? 0 : g;                  // clamp; junk masked later
            const float* gp = rowp + gc;
            unsigned la = lds_base + (unsigned)(i * 16);
            asm volatile("global_load_async_to_lds_b128 %0, %1, off"
                         :: "v"(la), "v"(gp) : "memory");
        }
        asm volatile("s_wait_asynccnt 0" ::: "memory");
    }
    __syncthreads();

    // ---- Phase B: per-thread sliding-window feature computation ----
    const int c0 = tile0 + tid * CPT;                  // first output t of this thread
    const float p0 = s_in[LDPAD];                      // price[row, tile0] (== price[0] on tile 0)

    float pl[NW];                                      // prices,  t = c0-21 .. c0+7
    float rl[NW];                                      // returns, same indexing
    const int lbase = tid * CPT + (LDPAD - HALO);      // s_in index of local j=0

#pragma unroll
    for (int j = 0; j < NW; ++j) {
        int t = c0 - HALO + j;
        float v = s_in[lbase + j];
        pl[j] = (t >= 0) ? v : 0.0f;                   // zero-padded before t=0
    }
    {
        float prev = __logf(pl[0] + EPSF);
        rl[0] = 0.0f;
#pragma unroll
        for (int j = 1; j < NW; ++j) {
            float cur = __logf(pl[j] + EPSF);
            int t = c0 - HALO + j;
            rl[j] = (t >= 1) ? (cur - prev) : 0.0f;    // r[0] = 0 (padded)
            prev = cur;
        }
    }

    // initialize accumulators for the first output (local j = 21, t = c0)
    float s5 = 0.f, q5 = 0.f, s10 = 0.f, q10 = 0.f, s20 = 0.f, q20 = 0.f;
    float ps5 = 0.f, ps20 = 0.f, bp = 0.f;
#pragma unroll
    for (int j = 2; j <= 21; ++j) { float r = rl[j]; s20 += r; q20 += r * r; ps20 += pl[j]; }
#pragma unroll
    for (int j = 12; j <= 21; ++j) { float r = rl[j]; s10 += r; q10 += r * r; }
#pragma unroll
    for (int j = 17; j <= 21; ++j) { float r = rl[j]; s5  += r; q5  += r * r; ps5 += pl[j]; }
#pragma unroll
    for (int j = 1; j <= 19; ++j)  bp += fabsf(rl[j]) * fabsf(rl[j + 1]);

    const float PI_2 = 1.5707963267948966f;
#pragma unroll
    for (int k = 0; k < CPT; ++k) {
        const int j = HALO + k;                        // local index of output t
        const int t = c0 + k;
        float tp1 = (float)(t + 1);
        float c5  = fminf(tp1, 5.0f);
        float c10 = fminf(tp1, 10.0f);
        float c20 = fminf(tp1, 20.0f);

        float f0 = std_from(s5,  q5,  c5);
        float f1 = std_from(s10, q10, c10);
        float f2 = std_from(s20, q20, c20);

        float sma5  = ps5  / c5;
        float sma20 = ps20 / c20;
        float f3 = (sma5 - sma20) / (sma20 + EPSF);

        float sh = (t >= 5) ? pl[j - 5] : p0;          // edge-replicated shift
        float f4 = (pl[j] - sh) / (sh + EPSF);

        float rv = f2 * f2;
        float bv = (t >= 20) ? bp * PI_2 : 0.0f;       // masked bipower
        float f5 = rv / (bv + EPSF);

        const int ob = (t - tile0) * NF;
        s_out[ob + 0] = f0; s_out[ob + 1] = f1; s_out[ob + 2] = f2;
        s_out[ob + 3] = f3; s_out[ob + 4] = f4; s_out[ob + 5] = f5;

        if (k + 1 < CPT) {                             // slide all windows by one step
            float rn  = rl[j + 1];
            float o5  = rl[j - 4], o10 = rl[j - 9], o20 = rl[j - 19];
            s5  += rn - o5;   q5  += rn * rn - o5  * o5;
            s10 += rn - o10;  q10 += rn * rn - o10 * o10;
            s20 += rn - o20;  q20 += rn * rn - o20 * o20;
            ps5  += pl[j + 1] - pl[j - 4];
            ps20 += pl[j + 1] - pl[j - 19];
            bp   += fabsf(rl[j - 1])  * fabsf(rl[j])
                  - fabsf(rl[j - 20]) * fabsf(rl[j - 19]);
        }
    }

    __syncthreads();

    // ---- Phase C: async DMA the contiguous 48KB feature tile LDS -> global ----
    {
        float* orow = out + ((long long)row * T_LEN + tile0) * NF;
        const unsigned lds_base = (unsigned)(uintptr_t)&s_out[0];
        const int OCH = TILE * NF / 4;                 // 3072 b128 chunks
        for (int i = tid; i < OCH; i += TPB) {
            float* gp = orow + i * 4;
            unsigned la = lds_base + (unsigned)(i * 16);
            asm volatile("global_store_async_from_lds_b128 %0, %1, off"
                         :: "v"(gp), "v"(la) : "memory");
        }
        asm volatile("s_wait_asynccnt 0" ::: "memory");
    }
}

extern "C" void kernel_launch(void* const* d_in, const int* in_sizes, int n_in,
                              void* d_out, int out_size, void* d_ws, size_t ws_size,
                              hipStream_t stream) {
    const float* prices = (const float*)d_in[0];
    float* out = (float*)d_out;
    const int T = T_LEN;
    const int B = in_sizes[0] / T;
    dim3 grid((unsigned)B, (unsigned)(T / TILE));
    RegimeFeatureExtractor_81106162418314_kernel<<<grid, dim3(TPB), 0, stream>>>(prices, out);
}